// Registration_45509473468730
// MI455X (gfx1250) — compile-verified
//
#include <hip/hip_runtime.h>
#include <cstdint>

// ---------------------------------------------------------------------------
// Multi-level rigid 3D registration for MI455X (gfx1250).
//
//  * Working set ~10MB -> resident in 192MB L2; eval is an L2-gather-latency
//    problem. No GEMM shape exists (only a broadcast 3x3 rotation), so WMMA
//    does not apply. CDNA5-specific paths used instead:
//      - global_load_async_to_lds_b128 (async copy engine, ASYNCcnt) to
//        stream the contiguous target volume into LDS, double buffered,
//        overlapped with the rotated trilinear gather on the VALU
//      - s_wait_asynccnt / s_wait_dscnt split counters for the ping-pong
//      - wave32 shuffle reductions
//  * All kernels templated on D (16/32/64) so index math lowers to
//    shifts/masks instead of division-by-invariant sequences.
//  * Serial optimizer sequenced by ~500 small launches (graph-capture safe).
// ---------------------------------------------------------------------------

#define B_       8
#define D0_      64
#define MAXB_    32          // max blocks per (batch,candidate) volume
#define PI_      3.14159265358979323846f

// ---------------- CDNA5 async-copy helpers (gfx1250 only) -------------------

__device__ __forceinline__ unsigned lds_byte_off(const void* generic_ptr) {
    // Low 32 bits of a generic LDS pointer are the LDS offset
    // (flat-LDS aperture).
    auto p3 = (__attribute__((address_space(3))) const char*)generic_ptr;
    return (unsigned)(unsigned long long)p3;
}

__device__ __forceinline__ void async_load_b128(unsigned lds_off,
                                                unsigned byte_off,
                                                const float* base) {
    // GVS mode: copies 16 bytes from base + byte_off to LDS at lds_off; tracked by ASYNCcnt.
    asm volatile("global_load_async_to_lds_b128 %0, %1, %2"
                 :
                 : "v"(lds_off), "v"(byte_off), "s"(base)
                 : "memory");
}

__device__ __forceinline__ void wait_async0() {
    asm volatile("s_wait_asynccnt 0" ::: "memory");
}
__device__ __forceinline__ void wait_async1() {
    asm volatile("s_wait_asynccnt 1" ::: "memory");
}
__device__ __forceinline__ void wait_ds0() {
    // Fence prior LDS reads/writes before the async engine overwrites a buffer.
    asm volatile("s_wait_dscnt 0" ::: "memory");
}

// ------ branchless trilinear sampling (map_coordinates order=1, cval=0) -----

template<int D>
__device__ __forceinline__ float sample_tri(const float* __restrict__ vol,
                                            float c0, float c1, float c2) {
    constexpr int DD = D * D;
    float f0 = floorf(c0), f1 = floorf(c1), f2 = floorf(c2);
    int   i0 = (int)f0, i1 = (int)f1, i2 = (int)f2;
    float w0 = c0 - f0, w1 = c1 - f1, w2 = c2 - f2;

    // per-axis corner weights, zeroed when that corner is out of bounds
    float A0 = ((unsigned)i0       < (unsigned)D) ? 1.f - w0 : 0.f;
    float B0 = ((unsigned)(i0 + 1) < (unsigned)D) ? w0       : 0.f;
    float A1 = ((unsigned)i1       < (unsigned)D) ? 1.f - w1 : 0.f;
    float B1 = ((unsigned)(i1 + 1) < (unsigned)D) ? w1       : 0.f;
    float A2 = ((unsigned)i2       < (unsigned)D) ? 1.f - w2 : 0.f;
    float B2 = ((unsigned)(i2 + 1) < (unsigned)D) ? w2       : 0.f;

    // clamped indices (weight already 0 when clamped corner was OOB)
    int a0 = min(max(i0, 0), D - 1),     b0 = min(max(i0 + 1, 0), D - 1);
    int a1 = min(max(i1, 0), D - 1),     b1 = min(max(i1 + 1, 0), D - 1);
    int a2 = min(max(i2, 0), D - 1),     b2 = min(max(i2 + 1, 0), D - 1);

    const float* p0 = vol + a0 * DD;
    const float* p1 = vol + b0 * DD;
    int r0 = a1 * D, r1 = b1 * D;
    // 8 unconditional gathers -> maximal memory-level parallelism vs L2
    float v000 = p0[r0 + a2], v001 = p0[r0 + b2];
    float v010 = p0[r1 + a2], v011 = p0[r1 + b2];
    float v100 = p1[r0 + a2], v101 = p1[r0 + b2];
    float v110 = p1[r1 + a2], v111 = p1[r1 + b2];

    float l00 = fmaf(B2, v001, A2 * v000);
    float l01 = fmaf(B2, v011, A2 * v010);
    float l10 = fmaf(B2, v101, A2 * v100);
    float l11 = fmaf(B2, v111, A2 * v110);
    float l0  = fmaf(B1, l01, A1 * l00);
    float l1  = fmaf(B1, l11, A1 * l10);
    return fmaf(B0, l1, A0 * l0);
}

// ---------------- pooling (mean over F^3 blocks of the 64^3 input) ----------

template<int DOUT, int F>
__global__ void reg_pool_kernel(const float* __restrict__ in, float* __restrict__ out) {
    constexpr int TOTAL = DOUT * DOUT * DOUT;          // power of two
    int idx = blockIdx.x * blockDim.x + threadIdx.x;
    if (idx >= B_ * TOTAL) return;
    int b = idx / TOTAL;                               // shift
    int r = idx & (TOTAL - 1);
    int z = r / (DOUT * DOUT);
    int y = (r / DOUT) & (DOUT - 1);
    int x = r & (DOUT - 1);
    const float* src = in + (size_t)b * (D0_ * D0_ * D0_);
    float s = 0.f;
#pragma unroll
    for (int dz = 0; dz < F; ++dz)
#pragma unroll
        for (int dy = 0; dy < F; ++dy)
#pragma unroll
            for (int dx = 0; dx < F; ++dx)
                s += src[((z * F + dz) * D0_ + (y * F + dy)) * D0_ + (x * F + dx)];
    out[idx] = s * (1.0f / (F * F * F));
}

// ---------------- tiny state kernels ---------------------------------------

__global__ void reg_init_theta(const float* __restrict__ theta_in, float* __restrict__ theta_deg) {
    int t = threadIdx.x;
    if (t < B_ * 6) {
        int c = t % 6;
        float inv_d2r = (c < 3) ? (180.0f / PI_) : 1.0f;   // theta / d2r
        theta_deg[t] = theta_in[t] * inv_d2r;
    }
}

__global__ void reg_level_init(float* __restrict__ buf, float* __restrict__ loss_all) {
    int t = threadIdx.x;
    if (t < B_ * 6) buf[t] = 0.f;
    if (t < B_)     loss_all[t] = 0.f;
}

__global__ void reg_step_init(int* __restrict__ active) {
    if (threadIdx.x < B_) active[threadIdx.x] = 1;
}

__global__ void reg_final(const float* __restrict__ theta_deg,
                          const float* __restrict__ loss_all,
                          float* __restrict__ out) {
    int t = threadIdx.x;
    if (t < B_ * 6) {
        int c = t % 6;
        float d2r = (c < 3) ? (PI_ / 180.0f) : 1.0f;
        out[t] = theta_deg[t] * d2r;                 // theta0 + (theta_deg*d2r - theta0)
    } else if (t < B_ * 6 + B_) {
        out[t] = loss_all[t - B_ * 6];
    }
}

// ---------------- fused warp + MSE partial-sum kernel -----------------------
// grid: (bpv, B*nCand); block 256. Each block handles 256*PTS contiguous
// points of one (batch,candidate) volume; exact tiling (D^3 == 256*PTS*bpv).
// Target streamed to LDS in 4KB groups (b128/lane) with ASYNCcnt ping-pong.

template<int D, int PTS>
__global__ void __launch_bounds__(256)
reg_eval_kernel(const float* __restrict__ src, const float* __restrict__ tgt,
                const float* __restrict__ theta_base, float* __restrict__ partials,
                float ss, int nCand) {
    constexpr int TOTAL  = D * D * D;
    constexpr int DD     = D * D;
    constexpr int GROUPS = PTS / 4;       // 1024 points (4KB) per group
    __shared__ float s_tgt[2][1024];
    __shared__ float warp_s[8];

    const int tid  = threadIdx.x;
    const int b    = blockIdx.y / nCand;
    const int c    = blockIdx.y - b * nCand;
    const int base = blockIdx.x * (256 * PTS);

    // candidate theta (degrees / translation units)
    float th[6];
#pragma unroll
    for (int i = 0; i < 6; ++i) th[i] = theta_base[b * 6 + i];
    if (c > 0) {
        if (c <= 6) th[c - 1] += ss;
        else        th[c - 7] -= ss;
    }
    const float ax = th[0] * (PI_ / 180.f);
    const float ay = th[1] * (PI_ / 180.f);
    const float az = th[2] * (PI_ / 180.f);
    const float t0 = th[3], t1 = th[4], t2 = th[5];
    float cx = cosf(ax), sx = sinf(ax);
    float cy = cosf(ay), sy = sinf(ay);
    float cz = cosf(az), sz = sinf(az);
    // R = Rz @ Ry @ Rx
    const float R00 = cz * cy, R01 = cz * sy * sx - sz * cx, R02 = cz * sy * cx + sz * sx;
    const float R10 = sz * cy, R11 = sz * sy * sx + cz * cx, R12 = sz * sy * cx - cz * sx;
    const float R20 = -sy,     R21 = cy * sx,                R22 = cy * cx;

    const float* __restrict__ srcb = src + (size_t)b * TOTAL;
    const float* __restrict__ tgtb = tgt + (size_t)b * TOTAL;

    constexpr float step = 2.0f / (float)(D - 1);
    constexpr float half = 0.5f * (float)(D - 1);

    // init LDS (liveness for the async writes) + per-thread 16B slot offsets
#pragma unroll
    for (int q = 0; q < 4; ++q) { s_tgt[0][tid * 4 + q] = 0.f; s_tgt[1][tid * 4 + q] = 0.f; }
    const unsigned lds_off[2] = { lds_byte_off(&s_tgt[0][tid * 4]),
                                  lds_byte_off(&s_tgt[1][tid * 4]) };

    // prologue: group 0 (thread t fetches elements 4t..4t+3 of the group)
    wait_ds0();
    async_load_b128(lds_off[0], (unsigned)((base + tid * 4) * 4), tgtb);

    float acc = 0.f;
    for (int g = 0; g < GROUPS; ++g) {
        if (g + 1 < GROUPS) {
            wait_ds0();      // prior reads of the buffer being overwritten done
            async_load_b128(lds_off[(g + 1) & 1],
                            (unsigned)((base + (g + 1) * 1024 + tid * 4) * 4), tgtb);
            wait_async1();   // group g landed (async loads complete in order)
        } else {
            wait_async0();
        }
#pragma unroll
        for (int sub = 0; sub < 4; ++sub) {
            float tv = s_tgt[g & 1][sub * 256 + tid];          // ds_load
            int pidx = base + g * 1024 + sub * 256 + tid;       // < TOTAL by tiling
            int i = pidx / DD;                                  // shifts (D pow2)
            int j = (pidx / D) & (D - 1);
            int k = pidx & (D - 1);
            float g0 = fmaf((float)i, step, -1.f);
            float g1 = fmaf((float)j, step, -1.f);
            float g2 = fmaf((float)k, step, -1.f);
            float p0 = R00 * g0 + R01 * g1 + R02 * g2 + t0;
            float p1 = R10 * g0 + R11 * g1 + R12 * g2 + t1;
            float p2 = R20 * g0 + R21 * g1 + R22 * g2 + t2;
            float v  = sample_tri<D>(srcb, (p0 + 1.f) * half,
                                           (p1 + 1.f) * half,
                                           (p2 + 1.f) * half);
            float dd = v - tv;
            acc = fmaf(dd, dd, acc);
        }
    }

    // wave32 reduction, then cross-wave via LDS
#pragma unroll
    for (int o = 16; o > 0; o >>= 1) acc += __shfl_down(acc, o, 32);
    int wid = tid >> 5, lane = tid & 31;
    if (lane == 0) warp_s[wid] = acc;
    __syncthreads();
    if (tid == 0) {
        float s = 0.f;
#pragma unroll
        for (int w = 0; w < 8; ++w) s += warp_s[w];
        partials[blockIdx.y * MAXB_ + blockIdx.x] = s;
    }
}

// ---------------- optimizer update 1: grad + momentum + theta_try -----------

__global__ void reg_u1_kernel(const float* __restrict__ partials,
                              const float* __restrict__ theta_deg,
                              float* __restrict__ buf, float* __restrict__ loss_all,
                              float* __restrict__ loss_theta, float* __restrict__ theta_try,
                              const int* __restrict__ active,
                              int bpv, float ss, int use_momentum, float invVol) {
    __shared__ float lbc[B_ * 13];
    for (int idx = threadIdx.x; idx < B_ * 13; idx += blockDim.x) {
        const float* p = partials + idx * MAXB_;
        float s = 0.f;
        for (int k = 0; k < bpv; ++k) s += p[k];
        lbc[idx] = s * invVol;
    }
    __syncthreads();
    if (threadIdx.x < B_) {
        int b = threadIdx.x;
        float loss = lbc[b * 13 + 0];
        int   act  = active[b];
        if (act) loss_all[b] = loss;
        loss_theta[b] = loss;
        float nb[6];
        float nrm2 = 0.f;
#pragma unroll
        for (int i = 0; i < 6; ++i) {
            float g = lbc[b * 13 + 1 + i] - lbc[b * 13 + 7 + i];  // lf[:6]-lf[6:]
            float o = buf[b * 6 + i];
            float u = use_momentum ? fmaf(o, 0.1f, g) : g;        // MOMENTUM=0.1
            float v = act ? u : o;                                // buf upd only if active
            nb[i] = v;
            nrm2 = fmaf(v, v, nrm2);
        }
        float scale = ss / (sqrtf(nrm2) + 1e-6f);
#pragma unroll
        for (int i = 0; i < 6; ++i) {
            buf[b * 6 + i] = nb[i];
            theta_try[b * 6 + i] = theta_deg[b * 6 + i] - scale * nb[i];
        }
    }
}

// ---------------- optimizer update 2: accept / reject -----------------------

__global__ void reg_u2_kernel(const float* __restrict__ partials,
                              float* __restrict__ theta_deg,
                              const float* __restrict__ theta_try,
                              const float* __restrict__ loss_theta,
                              int* __restrict__ active,
                              int bpv, float invVol) {
    if (threadIdx.x < B_) {
        int b = threadIdx.x;
        const float* p = partials + b * MAXB_;
        float s = 0.f;
        for (int k = 0; k < bpv; ++k) s += p[k];
        float loss_new = s * invVol;
        int na = active[b] && (loss_new + 1e-4f < loss_theta[b]);
        if (na) {
#pragma unroll
            for (int i = 0; i < 6; ++i) theta_deg[b * 6 + i] = theta_try[b * 6 + i];
        }
        active[b] = na;
    }
}

// ---------------- per-level host sequencing ---------------------------------

struct RegState {
    float *theta_deg, *buf, *loss_all, *loss_theta, *theta_try, *partials;
    int   *active;
};

template<int D, int PTS>
static void run_level(const float* sL, const float* tL, const RegState& st,
                      int level, bool& first_iter_of_level, hipStream_t stream) {
    constexpr int TOTAL = D * D * D;
    constexpr int BPV   = TOTAL / (256 * PTS);
    const float invVol  = 1.0f / (float)TOTAL;

    reg_level_init<<<1, 64, 0, stream>>>(st.buf, st.loss_all);
    first_iter_of_level = true;

    const float ss0 = 2.0f * (float)(1 << level);     // STEP_SIZE * 2^level
    for (int s = 0; s < 4; ++s) {                     // NUM_STEPS
        const float ss = ss0 / (float)(1 << s);
        reg_step_init<<<1, 32, 0, stream>>>(st.active);
        for (int it = 0; it < 10; ++it) {             // MAX_ITER
            reg_eval_kernel<D, PTS><<<dim3(BPV, B_ * 13), 256, 0, stream>>>(
                sL, tL, st.theta_deg, st.partials, ss, 13);
            reg_u1_kernel<<<1, 128, 0, stream>>>(
                st.partials, st.theta_deg, st.buf, st.loss_all, st.loss_theta,
                st.theta_try, st.active, BPV, ss,
                first_iter_of_level ? 0 : 1, invVol);
            first_iter_of_level = false;
            reg_eval_kernel<D, PTS><<<dim3(BPV, B_), 256, 0, stream>>>(
                sL, tL, st.theta_try, st.partials, ss, 1);
            reg_u2_kernel<<<1, 32, 0, stream>>>(
                st.partials, st.theta_deg, st.theta_try, st.loss_theta,
                st.active, BPV, invVol);
        }
    }
}

// ---------------- entry point -----------------------------------------------

extern "C" void kernel_launch(void* const* d_in, const int* in_sizes, int n_in,
                              void* d_out, int out_size, void* d_ws, size_t ws_size,
                              hipStream_t stream) {
    (void)in_sizes; (void)n_in; (void)out_size; (void)ws_size;

    const float* theta  = (const float*)d_in[0];   // (8,6)
    const float* source = (const float*)d_in[1];   // (8,1,64,64,64)
    const float* target = (const float*)d_in[2];   // (8,1,64,64,64)

    // workspace layout (floats); total ~2.4 MB
    float* w = (float*)d_ws;
    RegState st;
    st.theta_deg  = w;  w += B_ * 6;
    st.buf        = w;  w += B_ * 6;
    st.loss_all   = w;  w += B_;
    st.loss_theta = w;  w += B_;
    st.theta_try  = w;  w += B_ * 6;
    st.active     = (int*)w; w += B_;
    st.partials   = w;  w += B_ * 13 * MAXB_;
    float* src32  = w;  w += B_ * 32 * 32 * 32;
    float* tgt32  = w;  w += B_ * 32 * 32 * 32;
    float* src16  = w;  w += B_ * 16 * 16 * 16;
    float* tgt16  = w;  w += B_ * 16 * 16 * 16;

    reg_init_theta<<<1, 64, 0, stream>>>(theta, st.theta_deg);

    bool first;

    // level 2: 16^3
    {
        constexpr int N = (B_ * 16 * 16 * 16 + 255) / 256;
        reg_pool_kernel<16, 4><<<N, 256, 0, stream>>>(source, src16);
        reg_pool_kernel<16, 4><<<N, 256, 0, stream>>>(target, tgt16);
        run_level<16, 16>(src16, tgt16, st, 2, first, stream);
    }
    // level 1: 32^3
    {
        constexpr int N = (B_ * 32 * 32 * 32 + 255) / 256;
        reg_pool_kernel<32, 2><<<N, 256, 0, stream>>>(source, src32);
        reg_pool_kernel<32, 2><<<N, 256, 0, stream>>>(target, tgt32);
        run_level<32, 32>(src32, tgt32, st, 1, first, stream);
    }
    // level 0: 64^3
    run_level<64, 32>(source, target, st, 0, first, stream);

    reg_final<<<1, 64, 0, stream>>>(st.theta_deg, st.loss_all, (float*)d_out);
}